// ChartEncoder_69063074120431
// MI455X (gfx1250) — compile-verified
//
#include <hip/hip_runtime.h>
#include <math.h>

#define HH   100
#define HH2  200
#define LEN  64
#define NCELLS 2080            // 64*65/2
#define XS_STR 204             // LDS stride for X tile (16 x 200), conflict-free
#define H1_STR 116             // LDS stride for hidden tile (16 x 112), conflict-free

typedef float v2f __attribute__((ext_vector_type(2)));
typedef float v8f __attribute__((ext_vector_type(8)));

// D = A(16x4 f32) * B(4x16 f32) + C(16x16 f32)   -- native fp32 WMMA on gfx1250
__device__ __forceinline__ v8f wmma_f32(v2f a, v2f b, v8f c) {
  return __builtin_amdgcn_wmma_f32_16x16x4_f32(
      /*neg_a=*/false, a, /*neg_b=*/false, b,
      /*c_mod=*/(short)0, c, /*reuse_a=*/false, /*reuse_b=*/false);
}

__device__ __forceinline__ int off_of(int lvl) {
  int ll = LEN - lvl;
  return NCELLS - ll * (ll + 1) / 2;
}

__global__ void init_chart_kernel(const float* __restrict__ x,
                                  float* __restrict__ chart) {
  int idx = blockIdx.x * blockDim.x + threadIdx.x;
  if (idx < 128 * LEN * HH) {
    int b = idx / (LEN * HH);
    int r = idx - b * (LEN * HH);
    chart[(size_t)b * (NCELLS * HH) + r] = x[idx];
  }
}

__global__ void __launch_bounds__(256)
chart_level_kernel(float* __restrict__ chart,
                   const float* __restrict__ W1,
                   const float* __restrict__ b1,
                   const float* __restrict__ W2,
                   const float* __restrict__ b2,
                   const float* __restrict__ Ws,
                   const float* __restrict__ bs,
                   int level) {
  __shared__ float xs[16 * XS_STR];    // X tile: 16 rows x 200 (padded)
  __shared__ float h1s[16 * H1_STR];   // relu hidden: 16 rows x 112 (padded)
  __shared__ float wss[HH2];           // score weights
  __shared__ float smm[16], smd[16], sma[16], smw[16];  // online-softmax state

  const int tid  = threadIdx.x;
  const int wave = tid >> 5;
  const int lane = tid & 31;
  const int half = lane >> 4;
  const int l16  = lane & 15;

  const int j  = blockIdx.x;           // span index within this level
  const int b0 = blockIdx.y * 16;      // batch tile base
  const int Lv = LEN - level;
  const int Nv = level;
  const int olev = off_of(level);

  const int n0  = wave * 16;
  const int col = n0 + l16;            // this lane's output column
  const bool cvalid = (col < HH);

  // ---- stage Ws into LDS; init softmax state ----
  for (int idx = tid; idx < HH2; idx += 256) wss[idx] = Ws[idx];
  if (tid < 16) { smm[tid] = -__builtin_inff(); smd[tid] = 0.0f; }
  const float bsv = bs[0];

  // ---- preload this wave's weight columns into REGISTERS (loop-invariant) ----
  // B-operand layout for V_WMMA_F32_16X16X4_F32: lanes 0-15 hold K {0,1},
  // lanes 16-31 hold K {2,3} of each 4-deep chunk, at column n0+l16.
  v2f w1r[50];   // W1: K = 200 -> 50 chunks
  v2f w2r[25];   // W2: K = 100 -> 25 chunks
  float bc1 = 0.0f, bc2 = 0.0f;
  if (wave < 7) {
    bc1 = cvalid ? b1[col] : 0.0f;
    bc2 = cvalid ? b2[col] : 0.0f;
    #pragma unroll
    for (int kc = 0; kc < 50; ++kc) {
      int krow = kc * 4 + half * 2;
      w1r[kc].x = cvalid ? W1[krow * HH + col] : 0.0f;
      w1r[kc].y = cvalid ? W1[(krow + 1) * HH + col] : 0.0f;
    }
    #pragma unroll
    for (int kc = 0; kc < 25; ++kc) {
      int krow = kc * 4 + half * 2;
      w2r[kc].x = cvalid ? W2[krow * HH + col] : 0.0f;
      w2r[kc].y = cvalid ? W2[(krow + 1) * HH + col] : 0.0f;
    }
  }
  __syncthreads();

  v8f acc = {};  // softmax-weighted accumulator; wave w holds cols [16w,16w+16)

  // reference reshape semantics: flat index m = j*N + k; i = m/L, s = m%L
  int ii = (j * Nv) / Lv;
  int ss = (j * Nv) - ii * Lv;

  for (int k = 0; k < Nv; ++k) {
    const int lcell = off_of(ii) + ss;
    const int rcell = off_of(level - ii) - Lv + ss;
    // increment (i, s) for next split
    if (++ss == Lv) { ss = 0; ++ii; }

    // cooperative gather of X tile: 16 rows x [chart[lcell] | chart[rcell]]
    for (int idx = tid; idx < 16 * 50; idx += 256) {
      int row = idx / 50, q = idx - row * 50;
      const float* src = (q < 25)
        ? chart + ((size_t)(b0 + row) * NCELLS + lcell) * HH + q * 4
        : chart + ((size_t)(b0 + row) * NCELLS + rcell) * HH + (q - 25) * 4;
      float4 v = *(const float4*)src;
      float* dst = xs + row * XS_STR + ((q < 25) ? q * 4 : 100 + (q - 25) * 4);
      *(float4*)dst = v;
    }
    __syncthreads();

    if (wave < 7) {
      // GEMM1: X(16x200) @ W1 + b1, relu.  B operand comes from registers.
      v8f c;
      #pragma unroll
      for (int v = 0; v < 8; ++v) c[v] = bc1;
      const float* xrow = xs + l16 * XS_STR + half * 2;
      #pragma unroll
      for (int kc = 0; kc < 50; ++kc) {
        v2f a = *(const v2f*)(xrow + kc * 4);
        c = wmma_f32(a, w1r[kc], c);
      }
      #pragma unroll
      for (int v = 0; v < 8; ++v) {
        float hv = c[v] > 0.0f ? c[v] : 0.0f;
        h1s[(v + half * 8) * H1_STR + col] = hv;   // cols>=100 are exact zeros
      }
    } else if (lane < 16) {
      // wave 7: score s_raw[row] = X[row,:]·Ws + bs, then online-softmax update
      const float* xrow = xs + lane * XS_STR;
      float sacc = bsv;
      for (int q = 0; q < HH2; ++q) sacc += xrow[q] * wss[q];
      float m_old = smm[lane];
      float m_new = fmaxf(m_old, sacc);
      float alpha = expf(m_old - m_new);   // 0 on first split (m_old = -inf)
      float wgt   = expf(sacc - m_new);
      smd[lane] = smd[lane] * alpha + wgt;
      smm[lane] = m_new;
      sma[lane] = alpha;
      smw[lane] = wgt;
    }
    __syncthreads();

    if (wave < 7) {
      // GEMM2: relu(H1)(16x100) @ W2 + b2.  B operand from registers.
      v8f c;
      #pragma unroll
      for (int v = 0; v < 8; ++v) c[v] = bc2;
      const float* h1row = h1s + l16 * H1_STR + half * 2;
      #pragma unroll
      for (int kc = 0; kc < 25; ++kc) {
        v2f a = *(const v2f*)(h1row + kc * 4);
        c = wmma_f32(a, w2r[kc], c);
      }
      // online softmax accumulate: acc = acc*alpha[row] + w[row]*h
      #pragma unroll
      for (int v = 0; v < 8; ++v) {
        int row = v + half * 8;
        acc[v] = acc[v] * sma[row] + smw[row] * c[v];
      }
    }
    __syncthreads();
  }

  // finalize: hbar = acc / denom -> chart[b, off(level)+j, :]
  if (wave < 7 && cvalid) {
    const int ocell = olev + j;
    #pragma unroll
    for (int v = 0; v < 8; ++v) {
      int row = v + half * 8;
      chart[((size_t)(b0 + row) * NCELLS + ocell) * HH + col] = acc[v] / smd[row];
    }
  }
}

extern "C" void kernel_launch(void* const* d_in, const int* in_sizes, int n_in,
                              void* d_out, int out_size, void* d_ws, size_t ws_size,
                              hipStream_t stream) {
  (void)in_sizes; (void)n_in; (void)out_size; (void)d_ws; (void)ws_size;
  const float* x  = (const float*)d_in[0];
  const float* W1 = (const float*)d_in[1];
  const float* b1 = (const float*)d_in[2];
  const float* W2 = (const float*)d_in[3];
  const float* b2 = (const float*)d_in[4];
  const float* Ws = (const float*)d_in[5];
  const float* bs = (const float*)d_in[6];
  float* chart = (float*)d_out;

  // level 0: chart[:, :64] = x
  init_chart_kernel<<<(128 * LEN * HH + 255) / 256, 256, 0, stream>>>(x, chart);

  // levels 1..63, strictly ordered on `stream`
  for (int level = 1; level < LEN; ++level) {
    dim3 grid(LEN - level, 128 / 16);
    chart_level_kernel<<<grid, dim3(256), 0, stream>>>(
        chart, W1, b1, W2, b2, Ws, bs, level);
  }
}